// innerProdLoss_80496277062159
// MI455X (gfx1250) — compile-verified
//
#include <hip/hip_runtime.h>
#include <hip/hip_bf16.h>

// Problem constants (match reference)
#define W_IMG 96
#define H_IMG 72
#define NPTS  (W_IMG * H_IMG)          // 6912 = 432 * 16 exactly
#define NT    (NPTS / 16)              // 432 tiles per dimension
#define COL_CHUNKS 4
#define TILES_PER_CHUNK (NT / COL_CHUNKS)  // 108
#define DIST_COEF 0.1f

typedef float v2f __attribute__((ext_vector_type(2)));
typedef float v8f __attribute__((ext_vector_type(8)));

// ---------------------------------------------------------------------------
// Prep: back-project depths with invK (analytic), transform xyz2 by both
// poses, pack points as (x,y,z,|p|^2) and features as (fx,fy,fz,0) float4s.
// invK = [[0,0,1],[1/48,0,-1],[0,1/36,-1]]  (derived from CARLA K)
// ---------------------------------------------------------------------------
__global__ void ipl_prep(const float* __restrict__ depth1,
                         const float* __restrict__ depth2,
                         const float* __restrict__ pose,     // pose1_2, 4x4 row-major
                         const float* __restrict__ img1,     // (3,N)
                         const float* __restrict__ img2,     // (3,N)
                         const float* __restrict__ pose_gt,  // pose1_2_gt
                         float4* __restrict__ pts1,
                         float4* __restrict__ pts2t,
                         float4* __restrict__ pts2gt,
                         float4* __restrict__ f1p,
                         float4* __restrict__ f2p) {
    int n = blockIdx.x * blockDim.x + threadIdx.x;
    if (n >= NPTS) return;
    int u = n % W_IMG;
    int v = n / W_IMG;
    float g0 = 1.0f;
    float g1 = (float)u * (1.0f / 48.0f) - 1.0f;
    float g2 = (float)v * (1.0f / 36.0f) - 1.0f;

    float d1 = depth1[n];
    float x1 = d1 * g0, y1 = d1 * g1, z1 = d1 * g2;
    pts1[n] = make_float4(x1, y1, z1, x1 * x1 + y1 * y1 + z1 * z1);

    float d2 = depth2[n];
    float x2 = d2 * g0, y2 = d2 * g1, z2 = d2 * g2;

    // pose1_2 transform (rows 0..2 of 4x4, row-major)
    float tx = pose[0] * x2 + pose[1] * y2 + pose[2]  * z2 + pose[3];
    float ty = pose[4] * x2 + pose[5] * y2 + pose[6]  * z2 + pose[7];
    float tz = pose[8] * x2 + pose[9] * y2 + pose[10] * z2 + pose[11];
    pts2t[n] = make_float4(tx, ty, tz, tx * tx + ty * ty + tz * tz);

    float gx = pose_gt[0] * x2 + pose_gt[1] * y2 + pose_gt[2]  * z2 + pose_gt[3];
    float gy = pose_gt[4] * x2 + pose_gt[5] * y2 + pose_gt[6]  * z2 + pose_gt[7];
    float gz = pose_gt[8] * x2 + pose_gt[9] * y2 + pose_gt[10] * z2 + pose_gt[11];
    pts2gt[n] = make_float4(gx, gy, gz, gx * gx + gy * gy + gz * gz);

    f1p[n] = make_float4(img1[n], img1[NPTS + n], img1[2 * NPTS + n], 0.0f);
    f2p[n] = make_float4(img2[n], img2[NPTS + n], img2[2 * NPTS + n], 0.0f);
}

// ---------------------------------------------------------------------------
// Main: one wave per (pass, row_tile, col_chunk).
// Per 16x16 tile: 2x v_wmma_f32_16x16x4_f32 (distance trick + gram),
// then exp(-0.1*sq)*gram accumulated per lane.
//
// A-matrix 16x4 f32 layout (2 VGPRs/lane): lanes 0-15 = rows, v0/v1 hold
// K=0/1 (lanes 0-15) or K=2/3 (lanes 16-31). B mirrors with columns.
// C/D layout: column = lane&15, row = vgpr_i + 8*(lane>>4).
// ---------------------------------------------------------------------------
__global__ void __launch_bounds__(256) ipl_main(const float4* __restrict__ pts1,
                                                const float4* __restrict__ pts2t,
                                                const float4* __restrict__ pts2gt,
                                                const float4* __restrict__ f1p,
                                                const float4* __restrict__ f2p,
                                                double* __restrict__ accum) {
    const int gtid = blockIdx.x * blockDim.x + threadIdx.x;
    const int wave = gtid >> 5;
    const int lane = gtid & 31;

    int tmp = wave;
    const int chunk = tmp % COL_CHUNKS; tmp /= COL_CHUNKS;
    const int row_tile = tmp % NT;      tmp /= NT;
    const int pass = tmp;                         // 0..4 (grid sized exactly)
    if (pass >= 5) return;                        // uniform per wave

    const float4* pa; const float4* pb; const float4* fa; const float4* fb;
    switch (pass) {
        case 0:  pa = pts1;   pb = pts1;   fa = f1p; fb = f1p; break; // f1_f1
        case 1:  pa = pts2t;  pb = pts2t;  fa = f2p; fb = f2p; break; // f2_f2
        case 2:  pa = pts1;   pb = pts2t;  fa = f1p; fb = f2p; break; // f1_f2
        case 3:  pa = pts2gt; pb = pts2gt; fa = f2p; fb = f2p; break; // f2_f2_gt
        default: pa = pts1;   pb = pts2gt; fa = f1p; fb = f2p; break; // f1_f2_gt
    }

    const int half = lane >> 4;
    const int l15  = lane & 15;
    const int r    = row_tile * 16 + l15;

    const float4 par = pa[r];
    const float4 faf = fa[r];

    // Distance A channels: [-2x, -2y, -2z, |a|^2]
    v2f aD, aG;
    aD.x = half ? (-2.0f * par.z) : (-2.0f * par.x);
    aD.y = half ? par.w           : (-2.0f * par.y);
    // Gram A channels: [fx, fy, fz, 0]
    aG.x = half ? faf.z : faf.x;
    aG.y = half ? 0.0f  : faf.y;

    float acc = 0.0f;
    const int cbeg = chunk * TILES_PER_CHUNK;
    const int cend = cbeg + TILES_PER_CHUNK;
    for (int ct = cbeg; ct < cend; ++ct) {
        const int c = ct * 16 + l15;
        const float4 pbc = pb[c];
        const float4 fbc = fb[c];

        // Distance B channels: [x, y, z, 1]
        v2f bD, bG;
        bD.x = half ? pbc.z : pbc.x;
        bD.y = half ? 1.0f  : pbc.y;
        // Gram B channels: [fx, fy, fz, 0]
        bG.x = half ? fbc.z : fbc.x;
        bG.y = half ? 0.0f  : fbc.y;

        v8f zero8 = {};
        // P[i] = -2 a.b + |a|^2   (per row/col of the tile)
        v8f P = __builtin_amdgcn_wmma_f32_16x16x4_f32(
            false, aD, false, bD, (short)0, zero8, false, false);
        // G[i] = fa . fb
        v8f G = __builtin_amdgcn_wmma_f32_16x16x4_f32(
            false, aG, false, bG, (short)0, zero8, false, false);

        const float bn = pbc.w;  // |b|^2 for this lane's column
#pragma unroll
        for (int i = 0; i < 8; ++i) {
            float sq = P[i] + bn;                 // full squared distance
            acc += __expf(-DIST_COEF * sq) * G[i];
        }
    }

    // wave32 reduction
#pragma unroll
    for (int off = 16; off >= 1; off >>= 1)
        acc += __shfl_xor(acc, off, 32);
    if (lane == 0)
        atomicAdd(&accum[pass], (double)acc);
}

__global__ void ipl_finish(const double* __restrict__ accum,
                           float* __restrict__ out) {
    double s0 = accum[0], s1 = accum[1], s2 = accum[2], s3 = accum[3], s4 = accum[4];
    double inner_neg    = s0 + s1 - 2.0 * s2;
    double inner_neg_gt = s0 + s3 - 2.0 * s4;
    out[0] = (float)(inner_neg / inner_neg_gt);
}

// ---------------------------------------------------------------------------
extern "C" void kernel_launch(void* const* d_in, const int* in_sizes, int n_in,
                              void* d_out, int out_size, void* d_ws, size_t ws_size,
                              hipStream_t stream) {
    const float* depth1  = (const float*)d_in[0];
    const float* depth2  = (const float*)d_in[1];
    const float* pose    = (const float*)d_in[2];
    const float* img1    = (const float*)d_in[3];
    const float* img2    = (const float*)d_in[4];
    const float* pose_gt = (const float*)d_in[5];

    // Workspace layout (all 16B-aligned): 5 float4 arrays of NPTS, then 5 doubles.
    float4* pts1   = (float4*)d_ws;
    float4* pts2t  = pts1   + NPTS;
    float4* pts2gt = pts2t  + NPTS;
    float4* f1p    = pts2gt + NPTS;
    float4* f2p    = f1p    + NPTS;
    double* accum  = (double*)(f2p + NPTS);

    hipMemsetAsync(accum, 0, 5 * sizeof(double), stream);

    ipl_prep<<<(NPTS + 255) / 256, 256, 0, stream>>>(
        depth1, depth2, pose, img1, img2, pose_gt,
        pts1, pts2t, pts2gt, f1p, f2p);

    const int total_waves   = 5 * NT * COL_CHUNKS;   // 8640
    const int total_threads = total_waves * 32;      // 276480
    ipl_main<<<total_threads / 256, 256, 0, stream>>>(
        pts1, pts2t, pts2gt, f1p, f2p, accum);

    ipl_finish<<<1, 1, 0, stream>>>(accum, (float*)d_out);
}